// STFTLayer_78769700209024
// MI455X (gfx1250) — compile-verified
//
#include <hip/hip_runtime.h>
#include <math.h>

typedef __attribute__((ext_vector_type(2))) float v2f;
typedef __attribute__((ext_vector_type(8))) float v8f;

#define T_LEN            262144
#define NFRAMES          65535
#define NBINS            5
#define NBATCH           64
#define FRAMES_PER_BLOCK 128
#define FRAMES_PER_WAVE  16
#define ROW_PAD          132   // 128 + 4 floats: rows land on distinct bank groups

// fast atan2: |err| ~ few ulp, no special-case slow path (inputs are finite
// sums/differences of windowed Gaussians; only mx==0 is guarded)
__device__ __forceinline__ float fast_atan2f(float y, float x) {
    float ax = __builtin_fabsf(x), ay = __builtin_fabsf(y);
    float mx = fmaxf(ax, ay), mn = fminf(ax, ay);
    float t  = mn * __builtin_amdgcn_rcpf(mx);
    t = (mx == 0.f) ? 0.f : t;               // atan2(0,0) -> 0
    float s = t * t;
    float p =          2.6425264e-3f;        // ocml minimax coeffs, t in [0,1]
    p = fmaf(p, s, -1.5280068e-2f);
    p = fmaf(p, s,  4.1500056e-2f);
    p = fmaf(p, s, -7.4135043e-2f);
    p = fmaf(p, s,  1.0605230e-1f);
    p = fmaf(p, s, -1.4196968e-1f);
    p = fmaf(p, s,  1.9992290e-1f);
    p = fmaf(p, s, -3.3333063e-1f);
    float a = fmaf(t * s, p, t);             // atan(t)
    a = (ay > ax)  ? (1.57079632679489662f - a) : a;
    a = (x  < 0.f) ? (3.14159265358979323f - a) : a;
    return copysignf(a, y);
}

__global__ __launch_bounds__(256) void stft_wmma_kernel(
    const float* __restrict__ in, float* __restrict__ out_mag,
    float* __restrict__ out_ph)
{
    // [0]=Re, [1]=Im, rows padded to avoid 5-way bank conflicts on b128 stores
    __shared__ float lbuf[2][NBINS][ROW_PAD];

    const int tid    = threadIdx.x;
    const int lane   = tid & 31;
    const int wv     = tid >> 5;               // 8 waves per block
    const int b      = blockIdx.y;             // batch row
    const int fblock = blockIdx.x * FRAMES_PER_BLOCK;
    const int frame0 = fblock + wv * FRAMES_PER_WAVE;

    const int mn    = lane & 15;   // A: row M (frame); B: column N
    const int khalf = lane >> 4;   // selects K pair {0,1} vs {2,3}
    const int k0    = khalf * 2;

    // ---- A matrix: windowed frame samples (branchless window selects) ----
    int f  = frame0 + mn;
    int fc = f < NFRAMES ? f : (NFRAMES - 1);  // clamp: keep EXEC all-ones for WMMA
    const float* xb = in + (size_t)b * T_LEN + (size_t)fc * 4;

    v2f xlo = *(const v2f*)(xb + k0);          // samples k0, k0+1     (8B aligned)
    v2f xhi = *(const v2f*)(xb + 4 + k0);      // samples k0+4, k0+5   (8B aligned)
    // periodic Hann N=8: {0, .14645, .5, .85355, 1, .85355, .5, .14645}
    const bool h = (khalf != 0);
    v2f a01, a23;
    a01.x = xlo.x * (h ? 0.5f                  : 0.f);
    a01.y = xlo.y * (h ? 0.85355339059327373f  : 0.14644660940672624f);
    a23.x = xhi.x * (h ? 0.5f                  : 1.f);
    a23.y = xhi.y * (h ? 0.14644660940672624f  : 0.85355339059327373f);

    // ---- B matrix: DFT basis, branchless (v_cos/v_sin + selects) ----
    // n<5: Re basis cos(2*pi*k*n/8); 5<=n<10: Im basis -sin; n>=10: 0
    const bool  isim = (mn >= NBINS);
    const int   j    = isim ? (mn - NBINS) : mn;
    const bool  live = (mn < 2 * NBINS);
    v2f b01, b23;
    {
        const float PI4 = 0.78539816339744831f;
        float ang;
        ang = PI4 * (float)(( k0      * j) & 7);
        b01.x = live ? (isim ? -__sinf(ang) : __cosf(ang)) : 0.f;
        ang = PI4 * (float)(((k0 + 1) * j) & 7);
        b01.y = live ? (isim ? -__sinf(ang) : __cosf(ang)) : 0.f;
        ang = PI4 * (float)(((k0 + 4) * j) & 7);
        b23.x = live ? (isim ? -__sinf(ang) : __cosf(ang)) : 0.f;
        ang = PI4 * (float)(((k0 + 5) * j) & 7);
        b23.y = live ? (isim ? -__sinf(ang) : __cosf(ang)) : 0.f;
    }

    // ---- D = A(16x8) * B(8x16) as two chained 16x16x4 f32 WMMAs ----
    v8f c = {0.f, 0.f, 0.f, 0.f, 0.f, 0.f, 0.f, 0.f};
    c = __builtin_amdgcn_wmma_f32_16x16x4_f32(false, a01, false, b01,
                                              (short)0, c, false, false);
    c = __builtin_amdgcn_wmma_f32_16x16x4_f32(false, a23, false, b23,
                                              (short)0, c, false, false);

    // ---- stage Re/Im to LDS: c[r] for r=0..7 are consecutive local frames,
    //      so each live lane issues two ds_store_b128 ----
    if (live) {
        float* p = &lbuf[isim ? 1 : 0][j][wv * FRAMES_PER_WAVE + khalf * 8];
        float4 lo = { c[0], c[1], c[2], c[3] };
        float4 hi = { c[4], c[5], c[6], c[7] };
        *(float4*)(p)     = lo;
        *(float4*)(p + 4) = hi;
    }
    __syncthreads();

    // ---- all 256 threads compute mag/phase at full lane occupancy,
    //      stores are coalesced 128-float rows ----
    for (int i = tid; i < NBINS * FRAMES_PER_BLOCK; i += 256) {
        int bin = i >> 7;
        int fl  = i & 127;
        int fr  = fblock + fl;
        if (fr < NFRAMES) {
            float re = lbuf[0][bin][fl];
            float im = lbuf[1][bin][fl];
            float mag = __builtin_amdgcn_sqrtf(fmaf(re, re, im * im));
            float ph  = fast_atan2f(im, re);
            size_t o = ((size_t)b * NBINS + bin) * (size_t)NFRAMES + fr;
            __builtin_nontemporal_store(mag, out_mag + o);
            __builtin_nontemporal_store(ph,  out_ph  + o);
        }
    }
}

extern "C" void kernel_launch(void* const* d_in, const int* in_sizes, int n_in,
                              void* d_out, int out_size, void* d_ws, size_t ws_size,
                              hipStream_t stream) {
    (void)in_sizes; (void)n_in; (void)out_size; (void)d_ws; (void)ws_size;
    const float* in  = (const float*)d_in[0];
    float* out_mag   = (float*)d_out;
    float* out_ph    = out_mag + (size_t)NBATCH * NBINS * NFRAMES;

    dim3 grid((NFRAMES + FRAMES_PER_BLOCK - 1) / FRAMES_PER_BLOCK, NBATCH);
    stft_wmma_kernel<<<grid, 256, 0, stream>>>(in, out_mag, out_ph);
}